// SymFeats_25786983645301
// MI455X (gfx1250) — compile-verified
//
#include <hip/hip_runtime.h>

// ---------------- constants (ANI-1x rHCNO-5.2R_16-3.5A_a4-8) ----------------
#define WPB 8  // waves per block

typedef float v2f __attribute__((ext_vector_type(2)));
typedef float v8f __attribute__((ext_vector_type(8)));

constexpr int   kB = 64, kA = 48, kNS = 4;
constexpr int   kAEV = 384;                 // 64 radial + 320 angular
constexpr float kBohr2Ang = 0.52917721067f;
constexpr float kRcr = 5.2f, kEtaR = 16.0f;
constexpr float kRca = 3.5f, kEtaA = 8.0f;
constexpr float kPi  = 3.14159265358979323846f;

// ---------------- species passthrough (output 0, raw int bits) ----------------
__global__ void species_copy_kernel(const int* __restrict__ sp,
                                    int* __restrict__ out, int n) {
  int idx = blockIdx.x * blockDim.x + threadIdx.x;
  if (idx < n) out[idx] = sp[idx];
}

// ---------------- main AEV kernel: one wave32 per center atom ----------------
__global__ __launch_bounds__(256) void aev_kernel(const int* __restrict__ species,
                                                  const float* __restrict__ coords,
                                                  float* __restrict__ out) {
  // per-wave scratch (each wave uses its own slice; no cross-wave sync needed)
  __shared__ float sX[WPB][48], sY[WPB][48], sZ[WPB][48];   // compacted neighbors (fca>0)
  __shared__ float sD[WPB][48], sW[WPB][48];
  __shared__ int   sSp[WPB][48];
  __shared__ float sDall[WPB][48], sFcr[WPB][48];           // all-j (radial path)
  __shared__ int   sSpN[WPB][48];
  __shared__ int   sCnt[WPB];

  const int tid    = threadIdx.x;
  const int w      = tid >> 5;
  const int lane   = tid & 31;
  const int center = blockIdx.x * WPB + w;
  const int b      = center / kA;
  const int i      = center % kA;

  const int hi    = (lane >= 16) ? 1 : 0;
  const int fRow  = lane & 15;       // WMMA row (A) / column (B,D) for this lane

  // angular lane constants: tile0 feature f = fRow (a in {0,1}), tile1 f = fRow+16 (a+2)
  const int   zI    = fRow & 7;
  const int   aLo   = fRow >> 3;
  const float shfz  = (2 * zI + 1) * (kPi / 16.0f);
  const float czL   = __cosf(shfz);
  const float szL   = __sinf(shfz);
  const float shfaLo = 0.9f + 0.65f * (float)aLo;
  const float shfaHi = 0.9f + 0.65f * (float)(aLo + 2);

  if (lane == 0) sCnt[w] = 0;

  const float cxi = coords[(b * kA + i) * 3 + 0] * kBohr2Ang;
  const float cyi = coords[(b * kA + i) * 3 + 1] * kBohr2Ang;
  const float czi = coords[(b * kA + i) * 3 + 2] * kBohr2Ang;
  const int   spi = species[b * kA + i];
  const bool  vi  = (spi >= 0);

  // ---- pass over all j: distances, radial cutoff values, compacted angular list ----
  #pragma unroll
  for (int pass = 0; pass < 2; ++pass) {
    int j = lane + pass * 32;
    if (j < kA) {
      float dx = coords[(b * kA + j) * 3 + 0] * kBohr2Ang - cxi;
      float dy = coords[(b * kA + j) * 3 + 1] * kBohr2Ang - cyi;
      float dz = coords[(b * kA + j) * 3 + 2] * kBohr2Ang - czi;
      float d  = sqrtf(dx * dx + dy * dy + dz * dz + 1e-12f);
      int  spj = species[b * kA + j];
      bool ok  = (j != i) && vi && (spj >= 0);
      float fcr = (ok && d <= kRcr) ? (0.5f * __cosf(kPi * d * (1.0f / kRcr)) + 0.5f) : 0.0f;
      float fca = (ok && d <= kRca) ? (0.5f * __cosf(kPi * d * (1.0f / kRca)) + 0.5f) : 0.0f;
      sDall[w][j] = d;
      sFcr[w][j]  = fcr;
      sSpN[w][j]  = (spj >= 0) ? spj : 0;
      if (fca > 0.0f) {                       // cutoff-compacted neighbor list
        int slot = atomicAdd(&sCnt[w], 1);    // ds_add_rtn_u32; order-independent sum
        sX[w][slot] = dx; sY[w][slot] = dy; sZ[w][slot] = dz;
        sD[w][slot] = d;  sW[w][slot] = fca; sSp[w][slot] = spj;
      }
    }
  }

  // ---- radial sub-AEV via f32 WMMA: out[r,s] = sum_j rad[r,j] * onehot[j,s] ----
  // A (16x4 f32): lane -> row M=fRow (=shift r); K half {0,1} lanes<16, {2,3} lanes>=16,
  // VGPR index walks inside the half. B mirrored (col N=fRow = species).
  v8f accR = {0.f, 0.f, 0.f, 0.f, 0.f, 0.f, 0.f, 0.f};
  const int khalf = hi ? 2 : 0;
  const float shfr = 0.9f + 0.26875f * (float)fRow;
  #pragma unroll
  for (int kb = 0; kb < 12; ++kb) {           // K = 48 atoms in 12 steps of 4
    int j0 = kb * 4 + khalf;
    float d0 = sDall[w][j0],     f0 = sFcr[w][j0];
    float d1 = sDall[w][j0 + 1], f1 = sFcr[w][j0 + 1];
    float t0 = d0 - shfr, t1 = d1 - shfr;
    v2f av, bv;
    av.x = 0.25f * __expf(-kEtaR * t0 * t0) * f0;
    av.y = 0.25f * __expf(-kEtaR * t1 * t1) * f1;
    int s0 = sSpN[w][j0], s1 = sSpN[w][j0 + 1];
    bv.x = (s0 == fRow) ? 1.0f : 0.0f;
    bv.y = (s1 == fRow) ? 1.0f : 0.0f;
    accR = __builtin_amdgcn_wmma_f32_16x16x4_f32(false, av, false, bv,
                                                 (short)0, accR, false, false);
  }

  float* outC = out + (size_t)(kB * kA) + (size_t)center * kAEV;

  // D (16x16 f32): lane holds rows M=v+8*hi, col N=fRow. Cols 0..3 = species.
  if (fRow < kNS) {
    #pragma unroll
    for (int v = 0; v < 8; ++v)
      outC[fRow * 16 + (v + 8 * hi)] = accR[v];   // radial layout: s*16 + r
  }

  // ---- angular sub-AEV via f32 WMMA over groups of 4 neighbor pairs ----
  // ang[f, bucket] = sum_pairs val[f, pair] * onehot(bucket_pair)
  // tile0: features 0..15 (rows), tile1: features 16..31; cols = 10 species-pair buckets.
  const int n = sCnt[w];
  const int P = (n * (n - 1)) >> 1;           // unordered pairs among compacted neighbors
  const int G = (P + 3) >> 2;                 // groups of 4 (tail padded with weight 0)

  // per-lane pair evaluator: 2 feature values (aLo tile, aHi tile) + bucket index
  auto pair_vals = [&](int j, int k, float wmask,
                       float& vLo, float& vHi, int& bkt) {
    float xj = sX[w][j], yj = sY[w][j], zj = sZ[w][j];
    float dj = sD[w][j], fj = sW[w][j];
    float xk = sX[w][k], yk = sY[w][k], zk = sZ[w][k];
    float dk = sD[w][k], fk = sW[w][k];
    // cos(theta) clipped; sin(theta)=sqrt(1-c^2) since theta in [0,pi] -> no acos
    float ct = 0.95f * (xj * xk + yj * yk + zj * zk) / fmaxf(dj * dk, 1e-12f);
    ct = fminf(fmaxf(ct, -0.999999f), 0.999999f);
    float st = sqrtf(1.0f - ct * ct);
    float t = 0.5f * (1.0f + ct * czL + st * szL);       // (1+cos(theta-shfz))/2
    float p = t * t; p *= p; p *= p; p *= p; p *= p;     // t^32 (ZETA=32), shared by tiles
    float davg = 0.5f * (dj + dk);
    float uLo = davg - shfaLo, uHi = davg - shfaHi;
    float w2 = fj * fk * wmask;
    vLo = p * 2.0f * __expf(-kEtaA * uLo * uLo) * w2;
    vHi = p * 2.0f * __expf(-kEtaA * uHi * uHi) * w2;
    int sj = sSp[w][j], sk = sSp[w][k];
    int mn = min(sj, sk), mx = max(sj, sk);
    bkt = ((mn * (7 - mn)) >> 1) + mx;                   // triu pair index, 0..9
  };

  v8f c0 = {0.f, 0.f, 0.f, 0.f, 0.f, 0.f, 0.f, 0.f};
  v8f c1 = {0.f, 0.f, 0.f, 0.f, 0.f, 0.f, 0.f, 0.f};
  int jj = 0, kk = 1;                          // wave-uniform pair cursor
  for (int g = 0; g < G; ++g) {
    int j0 = jj, k0 = kk;  kk++; if (kk >= n) { jj++; kk = jj + 1; }
    int j1 = jj, k1 = kk;  kk++; if (kk >= n) { jj++; kk = jj + 1; }
    int j2 = jj, k2 = kk;  kk++; if (kk >= n) { jj++; kk = jj + 1; }
    int j3 = jj, k3 = kk;  kk++; if (kk >= n) { jj++; kk = jj + 1; }

    // lane owns K slots {khalf, khalf+1} of this 4-pair group
    int base = g * 4;
    int  jA = hi ? j2 : j0, kA2 = hi ? k2 : k0;
    int  jB = hi ? j3 : j1, kB2 = hi ? k3 : k1;
    bool vA = (base + khalf)     < P; if (!vA) { jA = 0; kA2 = 0; }
    bool vB = (base + khalf + 1) < P; if (!vB) { jB = 0; kB2 = 0; }

    float vAlo, vAhi, vBlo, vBhi; int bktA, bktB;
    pair_vals(jA, kA2, vA ? 1.0f : 0.0f, vAlo, vAhi, bktA);
    pair_vals(jB, kB2, vB ? 1.0f : 0.0f, vBlo, vBhi, bktB);

    v2f av0, av1, bv;
    av0.x = vAlo; av0.y = vBlo;                // tile0: features 0..15
    av1.x = vAhi; av1.y = vBhi;                // tile1: features 16..31
    bv.x = (bktA == fRow) ? 1.0f : 0.0f;       // shared one-hot B (same K pairs)
    bv.y = (bktB == fRow) ? 1.0f : 0.0f;
    c0 = __builtin_amdgcn_wmma_f32_16x16x4_f32(false, av0, false, bv,
                                               (short)0, c0, false, false);
    c1 = __builtin_amdgcn_wmma_f32_16x16x4_f32(false, av1, false, bv,
                                               (short)0, c1, false, false);
  }

  // D tiles: lane holds rows M=v+8*hi, col N=fRow (bucket). Valid buckets 0..9.
  // angular layout: 64 + bucket*32 + f,  f = a*8+z
  if (fRow < 10) {
    #pragma unroll
    for (int v = 0; v < 8; ++v) {
      int f = v + 8 * hi;
      outC[64 + fRow * 32 + f]      = c0[v];   // tile0: f in 0..15
      outC[64 + fRow * 32 + f + 16] = c1[v];   // tile1: f in 16..31
    }
  }
}

extern "C" void kernel_launch(void* const* d_in, const int* in_sizes, int n_in,
                              void* d_out, int out_size, void* d_ws, size_t ws_size,
                              hipStream_t stream) {
  (void)in_sizes; (void)n_in; (void)out_size; (void)d_ws; (void)ws_size;
  const int*   species = (const int*)d_in[0];
  const float* coords  = (const float*)d_in[1];
  float*       out     = (float*)d_out;

  // output 0: species passthrough (raw int32 bits into first B*A slots)
  species_copy_kernel<<<(kB * kA + 255) / 256, 256, 0, stream>>>(
      species, (int*)d_out, kB * kA);

  // output 1: AEV — one wave32 per center, 8 waves per block
  aev_kernel<<<(kB * kA) / WPB, 256, 0, stream>>>(species, coords, out);
}